// TestModel_38740605010029
// MI455X (gfx1250) — compile-verified
//
#include <hip/hip_runtime.h>
#include <stdint.h>

#define S 64
#define BATCH 16

typedef __attribute__((ext_vector_type(2))) float v2f;
typedef __attribute__((ext_vector_type(8))) float v8f;

// softsign via hardware reciprocal (v_rcp_f32, ~1 ulp): x / (1+|x|)
__device__ __forceinline__ float softsign_f(float x)
{
#if __has_builtin(__builtin_amdgcn_rcpf)
    return x * __builtin_amdgcn_rcpf(1.0f + __builtin_fabsf(x));
#else
    return x / (1.0f + __builtin_fabsf(x));
#endif
}
__device__ __forceinline__ int clamp63(int v) { v = v < 0 ? 0 : v; return v > 63 ? 63 : v; }

// ---- CDNA5 async global->LDS path (guarded: falls back to direct loads) ----
#if defined(__gfx1250__) && __has_builtin(__builtin_amdgcn_global_load_async_to_lds_b32)
#define ASYNC_LDS 1
typedef __attribute__((address_space(1))) int g_i32;
typedef __attribute__((address_space(3))) int l_i32;
__device__ __forceinline__ void async_b32(const float* gp, float* lp)
{
    __builtin_amdgcn_global_load_async_to_lds_b32(
        (g_i32*)(uintptr_t)gp,
        (l_i32*)(uint32_t)(uintptr_t)lp,
        0, 0);
}
template<int N> __device__ __forceinline__ void wait_async()
{
#if __has_builtin(__builtin_amdgcn_s_wait_asynccnt)
    __builtin_amdgcn_s_wait_asynccnt((short)N);
#else
    asm volatile("s_wait_asynccnt %0" :: "i"(N) : "memory");
#endif
}
#endif

// ---------------- First layer: 2 input channels -> 1, softsign ----------------
__global__ void k_first(const float* __restrict__ pre, const float* __restrict__ intra,
                        const float* __restrict__ w0, const float* __restrict__ b0,
                        float* __restrict__ out)
{
    int idx = blockIdx.x * blockDim.x + threadIdx.x;
    int x = idx & 63, y = (idx >> 6) & 63, z = (idx >> 12) & 63, b = idx >> 18;
    const float* pb = pre   + (size_t)b * S * S * S;
    const float* ib = intra + (size_t)b * S * S * S;
    float acc = b0[0];
    #pragma unroll
    for (int dz = -1; dz <= 1; ++dz) {
        int zz = clamp63(z + dz);
        #pragma unroll
        for (int dy = -1; dy <= 1; ++dy) {
            int yy = clamp63(y + dy);
            int rb = (zz * S + yy) * S;
            #pragma unroll
            for (int dx = -1; dx <= 1; ++dx) {
                int xx = clamp63(x + dx);
                int wi = ((dz + 1) * 3 + (dy + 1)) * 3 + (dx + 1);
                acc = fmaf(w0[wi],      pb[rb + xx], acc);
                acc = fmaf(w0[27 + wi], ib[rb + xx], acc);
            }
        }
    }
    out[idx] = softsign_f(acc);
}

// ---------------- Mid layer: 1->1 conv + softsign via fp32 WMMA --------------
// Multi-filter WMMA: A holds the nine (dz,dy) x-tap weight triples as matrix
// rows (duplicated across lane halves so D element r is combo-r's x-conv in
// every lane). B holds one input row Toeplitz-shifted into K=0..2 (K=3 weight
// is 0). One wave owns (batch, z, 16-wide x-tile), marches y with a 3-row
// rolling accumulator; edge padding = index clamps + extra adds at y=0/63.
// B operands are double-buffered through LDS with async loads when available.
__global__ void __launch_bounds__(256) k_mid(const float* __restrict__ in,
                                             const float* __restrict__ wl,
                                             const float* __restrict__ bl,
                                             float* __restrict__ out)
{
    int lane = threadIdx.x & 31;
    int wv   = threadIdx.x >> 5;
    int wid  = blockIdx.x * 8 + wv;          // 0..4095
    int tx = wid & 3;
    int z  = (wid >> 2) & 63;
    int b  = wid >> 8;
    int n    = lane & 15;
    int half = lane >> 4;
    int x0 = tx * 16;

    // B-matrix source columns for this lane (K rows 2*half, 2*half+1), x-clamped.
    int xs0 = clamp63(x0 + n - 1 + 2 * half);
    int xs1 = clamp63(x0 + n     + 2 * half);

    // A matrix: row m = n, combos c = n&7 in rows 0-7 and 8-15 (duplicated).
    // lane<16 supplies K=0,1 ; lane>=16 supplies K=2,3 (K=3 weight == 0).
    int c = n & 7;                            // combo 0..7 ; combo 8 via A2
    v2f A1, A2;
    A1.x = half ? wl[3 * c + 2] : wl[3 * c + 0];
    A1.y = half ? 0.0f          : wl[3 * c + 1];
    A2.x = half ? wl[26] : wl[24];            // combo 8 = (dz=+1, dy=+1)
    A2.y = half ? 0.0f   : wl[25];
    float bias = bl[0];

    const float* base  = in  + (size_t)b * S * S * S;
    float*       obase = out + (size_t)b * S * S * S + (size_t)z * S * S;

    // Six per-lane element pointers (3 clamped z-planes x 2 Toeplitz columns);
    // row yp is reached with a uniform +yp*S element offset.
    const float* g[6];
    #pragma unroll
    for (int dzi = 0; dzi < 3; ++dzi) {
        const float* p = base + (size_t)clamp63(z + dzi - 1) * S * S;
        g[2 * dzi + 0] = p + xs0;
        g[2 * dzi + 1] = p + xs1;
    }

    v8f zeroC = {0.f, 0.f, 0.f, 0.f, 0.f, 0.f, 0.f, 0.f};
    float accA = 0.f, accB = 0.f, accC = 0.f;   // out[y-1], out[y], out[y+1]

#ifdef ASYNC_LDS
    __shared__ float sm[8][2][6][32];
    #pragma unroll
    for (int j = 0; j < 6; ++j)                  // prologue: row 0 -> buf 0
        async_b32(g[j], &sm[wv][0][j][lane]);
#endif

    #pragma unroll 2
    for (int yp = 0; yp < S; ++yp) {
#ifdef ASYNC_LDS
        int cur = yp & 1, nxt = cur ^ 1;
        if (yp + 1 < S) {
            #pragma unroll
            for (int j = 0; j < 6; ++j)          // fetch next row while computing
                async_b32(g[j] + (size_t)(yp + 1) * S, &sm[wv][nxt][j][lane]);
            wait_async<6>();                     // current row's 6 are complete
        } else {
            wait_async<0>();
        }
#else
        if (yp + 1 < S)
            __builtin_prefetch(g[2] + (size_t)(yp + 1) * S, 0, 1);
#endif
        accC = 0.f;
        #pragma unroll
        for (int dzi = 0; dzi < 3; ++dzi) {
            v2f B;
#ifdef ASYNC_LDS
            B.x = sm[wv][cur][2 * dzi + 0][lane];
            B.y = sm[wv][cur][2 * dzi + 1][lane];
#else
            B.x = g[2 * dzi + 0][(size_t)yp * S];
            B.y = g[2 * dzi + 1][(size_t)yp * S];
#endif
            // D = A x B : element r of d == filter-r x-conv of this row (16 x's)
            v8f d = __builtin_amdgcn_wmma_f32_16x16x4_f32(
                        false, A1, false, B, (short)0, zeroC, false, false);
            float Fm1, F0, Fp1;                  // filters dy = -1, 0, +1
            if (dzi < 2) {
                Fm1 = d[3 * dzi]; F0 = d[3 * dzi + 1]; Fp1 = d[3 * dzi + 2];
            } else {
                v8f d2 = __builtin_amdgcn_wmma_f32_16x16x4_f32(
                            false, A2, false, B, (short)0, zeroC, false, false);
                Fm1 = d[6]; F0 = d[7]; Fp1 = d2[0];
            }
            accA += Fp1;                          // -> out[yp-1]
            accB += F0;                           // -> out[yp]
            accC += Fm1;                          // -> out[yp+1]
            if (yp == 0)  accB += Fm1;            // pad: out[0]  += F_{-1}(row 0)
            if (yp == 63) accB += Fp1;            // pad: out[63] += F_{+1}(row 63)
        }
        if (yp > 0 && half == 0)
            obase[(size_t)(yp - 1) * S + x0 + n] = softsign_f(accA + bias);
        accA = accB; accB = accC;
    }
    if (half == 0)
        obase[(size_t)63 * S + x0 + n] = softsign_f(accA + bias);
}

// ---------------- Last layer: 1->3 conv + softsign + downsamples -------------
__global__ void k_last(const float* __restrict__ h, const float* __restrict__ wX,
                       const float* __restrict__ bX, float* __restrict__ outAll)
{
    int idx = blockIdx.x * blockDim.x + threadIdx.x;
    int x = idx & 63, y = (idx >> 6) & 63, z = (idx >> 12) & 63, b = idx >> 18;
    const float* hb = h + (size_t)b * S * S * S;
    float a0 = bX[0], a1 = bX[1], a2 = bX[2];
    #pragma unroll
    for (int dz = -1; dz <= 1; ++dz) {
        int zz = clamp63(z + dz);
        #pragma unroll
        for (int dy = -1; dy <= 1; ++dy) {
            int yy = clamp63(y + dy);
            const float* row = hb + ((size_t)zz * S + yy) * S;
            #pragma unroll
            for (int dx = -1; dx <= 1; ++dx) {
                float v = row[clamp63(x + dx)];
                int wi = ((dz + 1) * 3 + (dy + 1)) * 3 + (dx + 1);
                a0 = fmaf(wX[wi],      v, a0);
                a1 = fmaf(wX[27 + wi], v, a1);
                a2 = fmaf(wX[54 + wi], v, a2);
            }
        }
    }
    float r0 = softsign_f(a0), r1 = softsign_f(a1), r2 = softsign_f(a2);

    const size_t v3 = (size_t)S * S * S;
    size_t spat = (size_t)z * S * S + (size_t)y * S + x;
    outAll[(size_t)(b * 3 + 0) * v3 + spat] = r0;
    outAll[(size_t)(b * 3 + 1) * v3 + spat] = r1;
    outAll[(size_t)(b * 3 + 2) * v3 + spat] = r2;

    const size_t off32 = (size_t)BATCH * 3 * v3;
    if (((x | y | z) & 1) == 0) {
        const size_t v = (size_t)32 * 32 * 32;
        size_t s = ((size_t)(z >> 1) * 32 + (y >> 1)) * 32 + (x >> 1);
        outAll[off32 + (size_t)(b * 3 + 0) * v + s] = r0;
        outAll[off32 + (size_t)(b * 3 + 1) * v + s] = r1;
        outAll[off32 + (size_t)(b * 3 + 2) * v + s] = r2;
    }
    const size_t off16 = off32 + (size_t)BATCH * 3 * 32 * 32 * 32;
    if (((x | y | z) & 3) == 0) {
        const size_t v = (size_t)16 * 16 * 16;
        size_t s = ((size_t)(z >> 2) * 16 + (y >> 2)) * 16 + (x >> 2);
        outAll[off16 + (size_t)(b * 3 + 0) * v + s] = r0;
        outAll[off16 + (size_t)(b * 3 + 1) * v + s] = r1;
        outAll[off16 + (size_t)(b * 3 + 2) * v + s] = r2;
    }
    const size_t off8 = off16 + (size_t)BATCH * 3 * 16 * 16 * 16;
    if (((x | y | z) & 7) == 0) {
        const size_t v = (size_t)8 * 8 * 8;
        size_t s = ((size_t)(z >> 3) * 8 + (y >> 3)) * 8 + (x >> 3);
        outAll[off8 + (size_t)(b * 3 + 0) * v + s] = r0;
        outAll[off8 + (size_t)(b * 3 + 1) * v + s] = r1;
        outAll[off8 + (size_t)(b * 3 + 2) * v + s] = r2;
    }
}

extern "C" void kernel_launch(void* const* d_in, const int* in_sizes, int n_in,
                              void* d_out, int out_size, void* d_ws, size_t ws_size,
                              hipStream_t stream)
{
    (void)in_sizes; (void)n_in; (void)out_size;
    const float* pre   = (const float*)d_in[0];
    const float* intra = (const float*)d_in[1];
    const float* w0    = (const float*)d_in[2];
    const float* b0    = (const float*)d_in[3];
    const float* wsW   = (const float*)d_in[4];   // (100,1,1,3,3,3) = 100*27
    const float* bsB   = (const float*)d_in[5];   // (100,1)
    const float* wX    = (const float*)d_in[6];   // (3,1,3,3,3)
    const float* bX    = (const float*)d_in[7];   // (3,)

    const size_t hElems = (size_t)BATCH * S * S * S;  // 4,194,304 floats = 16 MiB
    float *bufA, *bufB;
    if (ws_size >= 2 * hElems * sizeof(float)) {
        bufA = (float*)d_ws; bufB = bufA + hElems;
    } else if (ws_size >= hElems * sizeof(float)) {
        bufA = (float*)d_ws; bufB = (float*)d_out;    // res64 region as bounce
    } else {
        bufA = (float*)d_out; bufB = bufA + hElems;   // emergency fallback
    }

    dim3 blk(256);
    dim3 grdVox((unsigned)(hElems / 256));            // 16384 blocks
    k_first<<<grdVox, blk, 0, stream>>>(pre, intra, w0, b0, bufA);

    float* cur = bufA;
    float* nxt = bufB;
    for (int L = 0; L < 100; ++L) {
        // 4096 waves total: 512 blocks x 8 waves; each wave = (b, z, x-tile)
        k_mid<<<dim3(512), blk, 0, stream>>>(cur, wsW + 27 * L, bsB + L, nxt);
        float* t = cur; cur = nxt; nxt = t;
    }
    // after an even number of mid layers, cur == bufA (not aliasing d_out)
    k_last<<<grdVox, blk, 0, stream>>>(cur, wX, bX, (float*)d_out);
}